// DressedQuantumCircuit_87746181857410
// MI455X (gfx1250) — compile-verified
//
#include <hip/hip_runtime.h>
#include <hip/hip_bf16.h>

typedef __attribute__((ext_vector_type(2))) float v2f;
typedef __attribute__((ext_vector_type(8))) float v8f;

#define NQ      10
#define NSTATE  1024      // 2^NQ
#define DIN     512
#define NLAYER  4
#define QPAD    16        // q_in row padded to 16 floats

// ---------------------------------------------------------------------------
// Kernel 1: q_in = x @ pre_w.T + pre_b, FP32 WMMA (v_wmma_f32_16x16x4_f32).
// One wave32 per 16-row output tile. N padded 10 -> 16; padded lanes clamp
// their B-row index to row 9 (in-bounds, value irrelevant: those q_in columns
// are never read). This keeps the K-loop completely branch-free so EXEC stays
// all-ones and the loop is just 2x global_load_b64 + v_wmma per step.
// ---------------------------------------------------------------------------
__global__ __launch_bounds__(32) void pre_gemm_wmma(
    const float* __restrict__ x,      // (4096, 512)
    const float* __restrict__ pre_w,  // (10, 512)
    const float* __restrict__ pre_b,  // (10,)
    float* __restrict__ q_in)         // (4096, 16) padded
{
    const int lane  = threadIdx.x;          // 0..31
    const int m0    = blockIdx.x * 16;
    const int mrow  = m0 + (lane & 15);     // A: M = lane % 16
    const int khalf = (lane >> 4) * 2;      // A/B: K pair = {0,1} or {2,3}
    const int ncol  = lane & 15;            // B/D: N = lane % 16
    const int brow  = (ncol < NQ) ? ncol : (NQ - 1);  // clamp: stay in-bounds

    // Per-lane base pointers, 8-byte aligned (khalf even, rows 512 floats).
    const float2* ap = (const float2*)(x + (size_t)mrow * DIN + khalf);
    const float2* bp = (const float2*)(pre_w + (size_t)brow * DIN + khalf);

    v8f acc = {};
#pragma unroll 8
    for (int kb = 0; kb < DIN; kb += 4) {
        const float2 av = ap[kb >> 1];
        const float2 bv = bp[kb >> 1];
        v2f a; a.x = av.x; a.y = av.y;
        v2f b; b.x = bv.x; b.y = bv.y;
        acc = __builtin_amdgcn_wmma_f32_16x16x4_f32(
            /*neg_a=*/false, a, /*neg_b=*/false, b,
            /*c_mod=*/(short)0, acc, /*reuse_a=*/false, /*reuse_b=*/false);
    }

    const float bias  = pre_b[brow];             // garbage for ncol>=NQ: unused
    const int   mbase = m0 + ((lane >> 4) * 8);  // D: M = r + 8*(lane/16)
#pragma unroll
    for (int r = 0; r < 8; ++r) {
        q_in[(size_t)(mbase + r) * QPAD + ncol] = acc[r] + bias;
    }
}

// ---------------------------------------------------------------------------
// Kernel 2: statevector simulation, one wave32 per batch element,
// 1024-complex state resident in LDS (8 KB). Single-wave workgroup so
// __syncthreads() degenerates to a waitcnt fence (barrier -> S_NOP).
// 32 single-wave WGs/WGP * 8 KB = 256 KB < 320 KB LDS.
// ---------------------------------------------------------------------------

__device__ __forceinline__ void apply_ry(float2* st, int lane, int qubit, float theta)
{
    float s, c;
    __sincosf(theta * 0.5f, &s, &c);
    const int bp     = NQ - 1 - qubit;      // bit position (big-endian qubits)
    const int stride = 1 << bp;
#pragma unroll 4
    for (int t = 0; t < NSTATE / 2 / 32; ++t) {   // 16 pairs per lane
        const int p  = lane + 32 * t;             // pair index 0..511
        const int k0 = ((p >> bp) << (bp + 1)) | (p & (stride - 1));
        const int k1 = k0 + stride;
        const float2 a0 = st[k0];
        const float2 a1 = st[k1];
        st[k0] = make_float2(c * a0.x - s * a1.x, c * a0.y - s * a1.y);
        st[k1] = make_float2(s * a0.x + c * a1.x, s * a0.y + c * a1.y);
    }
    __syncthreads();
}

__device__ __forceinline__ void apply_rz(float2* st, int lane, int qubit, float theta)
{
    float s, c;
    __sincosf(theta * 0.5f, &s, &c);
    const int bp = NQ - 1 - qubit;
#pragma unroll 4
    for (int j = 0; j < NSTATE / 32; ++j) {       // elementwise, lane-private
        const int k = lane + 32 * j;
        // bit==0: * e^{-i th/2} = (c - i s); bit==1: * (c + i s)
        const float sg = ((k >> bp) & 1) ? s : -s;
        const float2 a = st[k];
        st[k] = make_float2(a.x * c - a.y * sg, a.x * sg + a.y * c);
    }
    __syncthreads();
}

__device__ __forceinline__ void apply_cnot(float2* st, int lane, int cq, int tq)
{
    const int cpos = NQ - 1 - cq;
    const int tpos = NQ - 1 - tq;
    const int lo = (cpos < tpos) ? cpos : tpos;
    const int hi = (cpos < tpos) ? tpos : cpos;
#pragma unroll 4
    for (int t = 0; t < NSTATE / 4 / 32; ++t) {   // 256 swaps, 8 per lane
        const int q    = lane + 32 * t;           // 8 free bits
        const int lowb = q & ((1 << lo) - 1);
        const int midb = (q >> lo) & ((1 << (hi - lo - 1)) - 1);
        const int hib  = q >> (hi - 1);
        int k = lowb | (midb << (lo + 1)) | (hib << (hi + 1));
        k |= (1 << cpos);                          // control bit = 1, target bit = 0
        const int k2 = k | (1 << tpos);
        const float2 a = st[k];
        const float2 b = st[k2];
        st[k]  = b;
        st[k2] = a;
    }
    __syncthreads();
}

__global__ __launch_bounds__(32) void circuit_kernel(
    const float* __restrict__ q_in,    // (4096, 16) padded
    const float* __restrict__ w_ry,    // (4, 10)
    const float* __restrict__ w_rz,    // (4, 10)
    const float* __restrict__ post_w,  // (1, 10)
    const float* __restrict__ post_b,  // (1,)
    float* __restrict__ out)           // (4096, 1)
{
    __shared__ float2 st[NSTATE];
    __shared__ float  red[32];

    const int lane = threadIdx.x;
    const int bidx = blockIdx.x;

    // |0...0>
#pragma unroll 4
    for (int j = 0; j < NSTATE / 32; ++j) {
        const int k = lane + 32 * j;
        st[k] = make_float2((k == 0) ? 1.0f : 0.0f, 0.0f);
    }
    __syncthreads();

    // Encoding layer: RY(q_in[b, i]) on each qubit
    for (int i = 0; i < NQ; ++i)
        apply_ry(st, lane, i, q_in[(size_t)bidx * QPAD + i]);

    // Variational layers
    for (int l = 0; l < NLAYER; ++l) {
        for (int i = 0; i < NQ; ++i) {
            apply_ry(st, lane, i, w_ry[l * NQ + i]);
            apply_rz(st, lane, i, w_rz[l * NQ + i]);
        }
        for (int i = 0; i < NQ; ++i)
            apply_cnot(st, lane, i, (i + 1) % NQ);
    }

    // Fused measurement:
    // out[b] = sum_k |amp_k|^2 * (sum_i post_w[i]*sign_i(k)) + post_b
    float wq[NQ];
#pragma unroll
    for (int i = 0; i < NQ; ++i) wq[i] = post_w[i];

    float accp = 0.0f;
#pragma unroll 2
    for (int j = 0; j < NSTATE / 32; ++j) {
        const int k = lane + 32 * j;
        const float2 a = st[k];
        const float p = a.x * a.x + a.y * a.y;
        float weff = 0.0f;
#pragma unroll
        for (int i = 0; i < NQ; ++i) {
            const float sgn = ((k >> (NQ - 1 - i)) & 1) ? -1.0f : 1.0f;
            weff += wq[i] * sgn;
        }
        accp += p * weff;
    }

    red[lane] = accp;
    __syncthreads();
    if (lane == 0) {
        float s = 0.0f;
#pragma unroll
        for (int t = 0; t < 32; ++t) s += red[t];
        out[bidx] = s + post_b[0];
    }
}

// ---------------------------------------------------------------------------
extern "C" void kernel_launch(void* const* d_in, const int* in_sizes, int n_in,
                              void* d_out, int out_size, void* d_ws, size_t ws_size,
                              hipStream_t stream)
{
    const float* x      = (const float*)d_in[0];  // (4096, 512)
    const float* pre_w  = (const float*)d_in[1];  // (10, 512)
    const float* pre_b  = (const float*)d_in[2];  // (10,)
    const float* post_w = (const float*)d_in[3];  // (1, 10)
    const float* post_b = (const float*)d_in[4];  // (1,)
    const float* w_ry   = (const float*)d_in[5];  // (4, 10)
    const float* w_rz   = (const float*)d_in[6];  // (4, 10)
    float* out = (float*)d_out;                   // (4096, 1)

    const int B = in_sizes[0] / DIN;              // 4096

    float* q_in = (float*)d_ws;                   // (B, 16) padded, 256 KB

    pre_gemm_wmma<<<dim3(B / 16), dim3(32), 0, stream>>>(x, pre_w, pre_b, q_in);
    circuit_kernel<<<dim3(B), dim3(32), 0, stream>>>(q_in, w_ry, w_rz,
                                                     post_w, post_b, out);
}